// QuantumGenerator_41420664602976
// MI455X (gfx1250) — compile-verified
//
#include <hip/hip_runtime.h>
#include <hip/hip_bf16.h>
#include <math.h>

typedef __attribute__((ext_vector_type(2))) float v2f;
typedef __attribute__((ext_vector_type(8))) float v8f;

#define LAYERS 6
#define NB     1024          // batch
#define FPB    4096          // floats per batch element at every stage boundary
#define BN_EPS 1e-5f

// ---------------------------------------------------------------------------
// Kernel 1: build per-channel 4x4 matrices M_c = (prod_l P * K_l)[0:4,0:4]
// channels: [0,256)=w1, [256,320)=w2, [320,336)=w3, [336,340)=w4
// ---------------------------------------------------------------------------
__global__ void build_M(const float* __restrict__ w1, const float* __restrict__ w2,
                        const float* __restrict__ w3, const float* __restrict__ w4,
                        float* __restrict__ Mbuf)
{
    int c = blockIdx.x * 64 + threadIdx.x;
    if (c >= 340) return;
    const float* wc;
    if      (c < 256) wc = w1 + (size_t)c        * (LAYERS*3);
    else if (c < 320) wc = w2 + (size_t)(c-256)  * (LAYERS*3);
    else if (c < 336) wc = w3 + (size_t)(c-320)  * (LAYERS*3);
    else              wc = w4 + (size_t)(c-336)  * (LAYERS*3);

    // CNOT_RING permutation: P[perm[b], b] = 1  =>  (P v)[perm[b]] = v[b]
    int perm[8];
    for (int b = 0; b < 8; ++b) {
        int b0=(b>>2)&1, b1=(b>>1)&1, b2=b&1;
        b1^=b0; b2^=b1; b0^=b2;
        perm[b] = (b0<<2)|(b1<<1)|b2;
    }

    float U[64], T[64];
    for (int i = 0; i < 64; ++i) U[i] = 0.f;
    for (int i = 0; i < 8;  ++i) U[i*8+i] = 1.f;

    for (int l = 0; l < LAYERS; ++l) {
        float R[3][2][2];
        for (int q = 0; q < 3; ++q) {
            float t = wc[l*3+q] * 0.5f;
            float cs = cosf(t), sn = sinf(t);
            R[q][0][0]=cs; R[q][0][1]=-sn; R[q][1][0]=sn; R[q][1][1]=cs;
        }
        // T = K @ U, K[i][k] = R0[i2][k2]*R1[i1][k1]*R2[i0][k0] (kron order)
        for (int i = 0; i < 8; ++i) {
            int i2=(i>>2)&1, i1=(i>>1)&1, i0=i&1;
            for (int j = 0; j < 8; ++j) {
                float acc = 0.f;
                for (int k = 0; k < 8; ++k) {
                    int k2=(k>>2)&1, k1=(k>>1)&1, k0=k&1;
                    float kv = R[0][i2][k2]*R[1][i1][k1]*R[2][i0][k0];
                    acc = fmaf(kv, U[k*8+j], acc);
                }
                T[i*8+j] = acc;
            }
        }
        // U' = P @ T : row perm[i] of U' = row i of T
        for (int i = 0; i < 8; ++i)
            for (int j = 0; j < 8; ++j)
                U[perm[i]*8+j] = T[i*8+j];
    }
    for (int m = 0; m < 4; ++m)
        for (int k = 0; k < 4; ++k)
            Mbuf[(size_t)c*16 + m*4 + k] = U[m*8+k];
}

// ---------------------------------------------------------------------------
// Kernel 2: one qcnn stage via V_WMMA_F32_16X16X4_F32 (templated so all
// addressing is shifts/masks; BN+ReLU fused on load for stages 2-4).
// One wave handles 16 patches of the same channel c:
//   D(16x16) = A(16x4: rows0-3 = M_c, rest 0) x B(4x16: columns = amp vectors)
// A layout (ISA 7.12.2): lane<16 -> {M[l][0],M[l][1]}, lane>=16 -> {M[l-16][2],M[l-16][3]}
// B layout (K = vgpr + 2*half): lane<16 -> {amp_n[0],amp_n[1]}, lane>=16 -> {amp_n[2],amp_n[3]}
// D: lanes 0-15, vgprs 0-3 hold psi[0..3] of patch n=lane.
// ---------------------------------------------------------------------------
template<int Cin, int Hin, bool BN>
__global__ void qcnn_stage(const float* __restrict__ in, float* __restrict__ out,
                           const float* __restrict__ Mbuf, const float* __restrict__ ab)
{
    constexpr int OW = Hin >> 1;
    constexpr int P  = OW * OW;                   // patches per (b,c)
    constexpr int BP = NB * P;

    const int lane = threadIdx.x & 31;
    const int wave = (blockIdx.x * blockDim.x + threadIdx.x) >> 5;
    const int n    = lane & 15;                   // patch slot / A-row
    const int half = lane >> 4;                   // 0: k=0,1   1: k=2,3

    int pg = wave * 16 + n;                       // global patch id, (c,b,p) order
    int c  = pg / BP;                             // power-of-two -> shift
    int r  = pg & (BP - 1);
    int b  = r / P;
    int p  = r & (P - 1);
    int oh = p / OW, ow = p & (OW - 1);

    // load this lane's two patch elements (one 2x1 row of the 2x2 window)
    const float* src = in + (((size_t)b * Cin + c) * Hin + (2*oh + half)) * Hin + 2*ow;
    float x0 = src[0], x1 = src[1];
    if constexpr (BN) {                           // fused BN + ReLU on the input tensor
        float ga = ab[2*c], be = ab[2*c+1];
        x0 = fmaxf(fmaf(ga, x0, be), 0.f);
        x1 = fmaxf(fmaf(ga, x1, be), 0.f);
    }

    // patch L2 norm across the lane pair (lane, lane^16)
    float s2 = x0*x0 + x1*x1;
    float so = __shfl_xor(s2, 16, 32);
    // 1/max(sqrt(s), 1e-9) == rsq(max(s, 1e-18)); v_rsq_f32 ~2^-22 rel err
    float inv = __builtin_amdgcn_rsqf(fmaxf(s2 + so, 1e-18f));

    v2f Bv; Bv.x = x0 * inv; Bv.y = x1 * inv;

    v2f Av; Av.x = 0.f; Av.y = 0.f;
    if (n < 4) {                                  // rows 0-3 = M_c; rows 4-15 zero
        const float* Mr = Mbuf + (size_t)c*16 + n*4 + half*2;
        Av.x = Mr[0]; Av.y = Mr[1];
    }

    v8f Cv = {};
    v8f Dv = __builtin_amdgcn_wmma_f32_16x16x4_f32(
        /*neg_a=*/false, Av, /*neg_b=*/false, Bv,
        /*c_mod=*/(short)0, Cv, /*reuse_a=*/false, /*reuse_b=*/false);

    if (half == 0) {                              // lanes 0-15: psi of patch n in Dv[0..3]
        float q0 = Dv[0]*Dv[0], q1 = Dv[1]*Dv[1];
        float q2 = Dv[2]*Dv[2], q3 = Dv[3]*Dv[3];
        float invs = __builtin_amdgcn_rcpf(fmaxf(q0+q1+q2+q3, 1e-9f));
        float4 o;
        o.x = q0*invs; o.y = q1*invs; o.z = q2*invs; o.w = q3*invs;
        // flat per-batch index 4*(c*P+p) == natural (Cout,S,S) layout of the reshape
        *(float4*)(out + (size_t)b * FPB + ((size_t)c * P + p) * 4) = o;
    }
}

// ---------------------------------------------------------------------------
// Kernel 3: per-channel sum / sum-of-squares over (B,H,W). Grid = 1024 blocks.
// ---------------------------------------------------------------------------
__global__ void bn_reduce(const float* __restrict__ s, float* __restrict__ accum,
                          int C, int S)
{
    const int bpc = 1024 / C;                 // blocks per channel
    const int c   = blockIdx.x / bpc;
    const int q   = blockIdx.x - c * bpc;
    const int bPerBlock = NB / bpc;
    const int HW  = S * S;

    float sum = 0.f, ssq = 0.f;
    for (int b = q * bPerBlock; b < (q+1) * bPerBlock; ++b) {
        const float4* base = (const float4*)(s + ((size_t)b * C + c) * HW);
        for (int i = threadIdx.x; i < (HW >> 2); i += 256) {
            float4 x = base[i];
            sum += x.x + x.y + x.z + x.w;
            ssq = fmaf(x.x, x.x, fmaf(x.y, x.y, fmaf(x.z, x.z, fmaf(x.w, x.w, ssq))));
        }
    }
    __shared__ float ls[256], lss[256];
    ls[threadIdx.x] = sum; lss[threadIdx.x] = ssq;
    __syncthreads();
    for (int o = 128; o > 0; o >>= 1) {
        if (threadIdx.x < o) { ls[threadIdx.x] += ls[threadIdx.x+o]; lss[threadIdx.x] += lss[threadIdx.x+o]; }
        __syncthreads();
    }
    if (threadIdx.x == 0) {
        atomicAdd(&accum[2*c],   ls[0]);
        atomicAdd(&accum[2*c+1], lss[0]);
    }
}

// ---------------------------------------------------------------------------
// Kernel 4: finalize BN -> affine (a,b) per channel:  y = a*x + b
// ---------------------------------------------------------------------------
__global__ void bn_finalize(const float* __restrict__ accum,
                            const float* __restrict__ gamma, const float* __restrict__ beta,
                            float* __restrict__ ab, int C, float invN)
{
    int c = threadIdx.x;
    if (c >= C) return;
    float mean = accum[2*c] * invN;
    float var  = fmaf(-mean, mean, accum[2*c+1] * invN);
    float a = gamma[c] * rsqrtf(var + BN_EPS);
    ab[2*c]   = a;
    ab[2*c+1] = fmaf(-mean, a, beta[c]);
}

// ---------------------------------------------------------------------------
// Kernel 5: final outputs: tanh(s4_bn) and s4_bn (s4 already in d_out)
// ---------------------------------------------------------------------------
__global__ void final_out(const float* __restrict__ s4, const float* __restrict__ abf,
                          float* __restrict__ out_tanh, float* __restrict__ out_bn)
{
    int i = blockIdx.x * 256 + threadIdx.x;
    float bn = fmaf(abf[0], s4[i], abf[1]);
    out_bn[i]   = bn;
    out_tanh[i] = tanhf(bn);
}

// ---------------------------------------------------------------------------
extern "C" void kernel_launch(void* const* d_in, const int* in_sizes, int n_in,
                              void* d_out, int out_size, void* d_ws, size_t ws_size,
                              hipStream_t stream)
{
    const float* z   = (const float*)d_in[0];
    const float* w1  = (const float*)d_in[1];
    const float* w2  = (const float*)d_in[2];
    const float* w3  = (const float*)d_in[3];
    const float* w4  = (const float*)d_in[4];
    const float* g2  = (const float*)d_in[5];
    const float* b2  = (const float*)d_in[6];
    const float* g3  = (const float*)d_in[7];
    const float* b3  = (const float*)d_in[8];
    const float* g4  = (const float*)d_in[9];
    const float* b4  = (const float*)d_in[10];
    const float* gf  = (const float*)d_in[11];
    const float* bf  = (const float*)d_in[12];
    float* out = (float*)d_out;

    const size_t T = (size_t)NB * FPB;            // 4,194,304 elems per stage tensor
    float* ws   = (float*)d_ws;
    float* Mbuf = ws;                             // 340*16 = 5440
    float* s1   = ws + 5440;
    float* s2   = s1 + T;
    float* s3   = s2 + T;
    float* acc1 = s3 + T;                         // 128
    float* acc2 = acc1 + 128;                     // 32
    float* acc3 = acc2 + 32;                      // 8
    float* acc4 = acc3 + 8;                       // 2
    float* ab2  = acc4 + 2;                       // 128
    float* ab3  = ab2 + 128;                      // 32
    float* ab4  = ab3 + 32;                       // 8
    float* abf  = ab4 + 8;                        // 2

    float* o_tanh = out;
    float* o_s4   = out + T;
    float* o_bn   = out + 2*T;

    hipMemsetAsync(acc1, 0, (128+32+8+2) * sizeof(float), stream);

    build_M<<<6, 64, 0, stream>>>(w1, w2, w3, w4, Mbuf);

    // stage 1: z (1024,256,4,4) -> s1 (1024,64,8,8)
    qcnn_stage<256, 4, false><<<8192, 256, 0, stream>>>(z, s1, Mbuf, nullptr);
    bn_reduce<<<1024, 256, 0, stream>>>(s1, acc1, 64, 8);
    bn_finalize<<<1, 64, 0, stream>>>(acc1, g2, b2, ab2, 64, 1.f/(1024.f*64.f));

    // stage 2: s1 -> s2 (1024,16,16,16), BN+ReLU fused on load
    qcnn_stage<64, 8, true><<<8192, 256, 0, stream>>>(s1, s2, Mbuf + 256*16, ab2);
    bn_reduce<<<1024, 256, 0, stream>>>(s2, acc2, 16, 16);
    bn_finalize<<<1, 64, 0, stream>>>(acc2, g3, b3, ab3, 16, 1.f/(1024.f*256.f));

    // stage 3: s2 -> s3 (1024,4,32,32)
    qcnn_stage<16, 16, true><<<8192, 256, 0, stream>>>(s2, s3, Mbuf + 320*16, ab3);
    bn_reduce<<<1024, 256, 0, stream>>>(s3, acc3, 4, 32);
    bn_finalize<<<1, 64, 0, stream>>>(acc3, g4, b4, ab4, 4, 1.f/(1024.f*1024.f));

    // stage 4: s3 -> s4 (1024,1,64,64), written straight into d_out slot 1
    qcnn_stage<4, 32, true><<<8192, 256, 0, stream>>>(s3, o_s4, Mbuf + 336*16, ab4);
    bn_reduce<<<1024, 256, 0, stream>>>(o_s4, acc4, 1, 64);
    bn_finalize<<<1, 64, 0, stream>>>(acc4, gf, bf, abf, 1, 1.f/(1024.f*4096.f));

    // outputs: tanh(s4_bn), (s4 already placed), s4_bn
    final_out<<<16384, 256, 0, stream>>>(o_s4, abf, o_tanh, o_bn);
}